// CharToVectorLayer1_26233660244450
// MI455X (gfx1250) — compile-verified
//
#include <hip/hip_runtime.h>

// Embedding gather: out[row, 0:64] = table[x[row], 0:64]
//   x:     524,288 int ids   (256*256*8)
//   table: 8000 x 64 f32     (2 MB -> stays resident in MI455X's 192 MB L2)
//   out:   524,288 x 64 f32  (128 MiB, write-once stream)
//
// Pure bandwidth op (0 FLOPs). Roofline floor ~5.7 us at 23.3 TB/s.
// Strategy:
//   * 16 lanes per embedding row, one float4 (b128) per lane
//     -> half-wave = 256 contiguous bytes, wave32 = 512 contiguous bytes
//   * table reads: regular temporal loads (we WANT them cached in L2)
//   * output stores: non-temporal hint (TH_STORE_NT) so the 128 MiB stream
//     does not evict the 2 MB table from cache

typedef float v4f __attribute__((ext_vector_type(4)));

__global__ __launch_bounds__(256) void char2vec_gather_kernel(
    const int* __restrict__ x,
    const float* __restrict__ table,
    float* __restrict__ out,
    int n_rows)
{
    const int tid  = blockIdx.x * blockDim.x + threadIdx.x;
    const int row  = tid >> 4;          // which id / output row (16 lanes per row)
    const int lane = tid & 15;          // which float4 within the 64-float row

    if (row >= n_rows) return;

    // All 16 lanes of a half-wave read the same dword -> coalesces to one
    // request; id stream is only 2 MiB, stays hot.
    const int id = x[row];

    // Gather 16 B from the (L2-resident) table row.
    const v4f* __restrict__ src =
        reinterpret_cast<const v4f*>(table + (size_t)id * 64);
    v4f v = src[lane];

    // Streaming store of the 128 MiB output with non-temporal cache hint
    // (emits global_store_b128 ... th:TH_STORE_NT on gfx1250) so it does not
    // thrash the L2 lines holding the embedding table.
    v4f* __restrict__ dst =
        reinterpret_cast<v4f*>(out + (size_t)row * 64);
    __builtin_nontemporal_store(v, dst + lane);
}

extern "C" void kernel_launch(void* const* d_in, const int* in_sizes, int n_in,
                              void* d_out, int out_size, void* d_ws, size_t ws_size,
                              hipStream_t stream) {
    const int*   x     = (const int*)d_in[0];    // ids (harness: integer -> int32)
    const float* table = (const float*)d_in[1];  // [8000, 64] f32
    float*       out   = (float*)d_out;          // [n_rows, 64] f32

    const int n_rows = in_sizes[0];              // 524,288 ids
    const long long total_threads = (long long)n_rows * 16;

    const int threads = 256;                     // 8 wave32 per block
    const int blocks  = (int)((total_threads + threads - 1) / threads);

    char2vec_gather_kernel<<<blocks, threads, 0, stream>>>(x, table, out, n_rows);
}